// GATSummarizer_2241972928672
// MI455X (gfx1250) — compile-verified
//
#include <hip/hip_runtime.h>
#include <hip/hip_bf16.h>

// ---------------------------------------------------------------------------
// GAT summarizer for MI455X (gfx1250, wave32).
// GEMMs use V_WMMA_F32_16X16X4_F32 (exact fp32 matrix path, one wave per
// 16x16 tile). Edge softmax/scatter uses L2-resident fp32 atomics (node
// arrays are 19MB << 192MB L2).
// ---------------------------------------------------------------------------

#define N_NODES 50000
#define N_EDGES 800000
#define IN_CH   128
#define HID     48
#define HEADS   2
#define HH      96
#define N_GRAPHS 64
#define E_TOT   (N_EDGES + N_NODES)   // with self loops
#define NEG_SLOPE 0.2f

typedef float v2f __attribute__((ext_vector_type(2)));
typedef float v8f __attribute__((ext_vector_type(8)));

// ---- order-preserving float <-> uint for atomicMax-based segment max ------
__device__ __forceinline__ unsigned f2ord(float f) {
    unsigned u = __float_as_uint(f);
    return (u & 0x80000000u) ? ~u : (u | 0x80000000u);
}
__device__ __forceinline__ float ord2f(unsigned u) {
    u = (u & 0x80000000u) ? (u & 0x7FFFFFFFu) : ~u;
    return __uint_as_float(u);
}

// ---------------------------------------------------------------------------
// GEMM: C[M,N] = A[M,K] @ B[K,N], fp32, one wave per 16x16 C tile.
// Requires M%16==0, N%16==0, K%4==0 (true here: M=50000, N=96, K=128/96).
// ---------------------------------------------------------------------------
__global__ void gemm_wmma_f32(const float* __restrict__ A,
                              const float* __restrict__ B,
                              float* __restrict__ C, int M, int N, int K) {
    const int lane = threadIdx.x & 31;
    const int half = lane >> 4;      // 0: lanes 0-15, 1: lanes 16-31
    const int r    = lane & 15;      // row (A) / col (B,C) within tile
    const int m0   = blockIdx.x << 4;
    const int n0   = blockIdx.y << 4;

    v8f acc = {0.f, 0.f, 0.f, 0.f, 0.f, 0.f, 0.f, 0.f};

#if defined(__AMDGCN__) && __has_builtin(__builtin_amdgcn_wmma_f32_16x16x4_f32)
    const float* arow = A + (size_t)(m0 + r) * K;
    for (int k0 = 0; k0 < K; k0 += 4) {
        const int ka = k0 + 2 * half;
        // A fragment: lane holds A[r, ka], A[r, ka+1]  (ISA 16x4 f32 layout)
        v2f a;
        a.x = arow[ka];
        a.y = arow[ka + 1];
        // B fragment: lane holds B[ka, n0+r], B[ka+1, n0+r]
        v2f b;
        b.x = B[(size_t)ka * N + n0 + r];
        b.y = B[(size_t)(ka + 1) * N + n0 + r];
        acc = __builtin_amdgcn_wmma_f32_16x16x4_f32(
            /*neg_a=*/false, a, /*neg_b=*/false, b,
            /*c_mod=*/(short)0, acc, /*reuse_a=*/false, /*reuse_b=*/false);
    }
#else
    // Fallback (host stub pass / missing builtin): plain FMA, same layout.
    for (int v = 0; v < 8; ++v) {
        float s = 0.f;
        const int m = m0 + v + 8 * half;
        for (int k = 0; k < K; ++k)
            s += A[(size_t)m * K + k] * B[(size_t)k * N + n0 + r];
        acc[v] = s;
    }
#endif
    // C/D layout: VGPR v, lane -> C[m0 + v + 8*half, n0 + r]
    for (int v = 0; v < 8; ++v)
        C[(size_t)(m0 + v + 8 * half) * N + (n0 + r)] = acc[v];
}

// ---------------------------------------------------------------------------
// Per-node attention logits: es[i,h] = <h[i,h,:], a_src[h,:]>, same for ed.
// ---------------------------------------------------------------------------
__global__ void node_scores(const float* __restrict__ h,
                            const float* __restrict__ a_s,
                            const float* __restrict__ a_d,
                            float* __restrict__ es, float* __restrict__ ed,
                            int n, int heads, int C) {
    int idx = blockIdx.x * blockDim.x + threadIdx.x;
    if (idx >= n * heads) return;
    int node = idx / heads, hd = idx - node * heads;
    const float* hv = h + (size_t)node * heads * C + (size_t)hd * C;
    float s = 0.f, d = 0.f;
    for (int c = 0; c < C; ++c) {
        float v = hv[c];
        s += v * a_s[hd * C + c];
        d += v * a_d[hd * C + c];
    }
    es[idx] = s;
    ed[idx] = d;
}

__device__ __forceinline__ void edge_sd(int e, const int* srcI, const int* dstI,
                                        int& s, int& d) {
    if (e < N_EDGES) { s = srcI[e]; d = dstI[e]; }
    else             { s = e - N_EDGES; d = s; }   // self loop
}

// ---- pass 2: segment max of leaky_relu(es[src]+ed[dst]) over dst ----------
__global__ void edge_max(const int* __restrict__ srcI, const int* __restrict__ dstI,
                         const float* __restrict__ es, const float* __restrict__ ed,
                         unsigned* __restrict__ mbits, int heads) {
    int idx = blockIdx.x * blockDim.x + threadIdx.x;
    if (idx >= E_TOT * heads) return;
    int e = idx / heads, hd = idx - e * heads;
    int s, d; edge_sd(e, srcI, dstI, s, d);
    float v = es[s * heads + hd] + ed[d * heads + hd];
    v = v > 0.f ? v : NEG_SLOPE * v;
    atomicMax(&mbits[d * heads + hd], f2ord(v));
}

// ---- pass 3: p = exp(e - m[dst]); denom[dst] += p -------------------------
__global__ void edge_expsum(const int* __restrict__ srcI, const int* __restrict__ dstI,
                            const float* __restrict__ es, const float* __restrict__ ed,
                            const unsigned* __restrict__ mbits,
                            float* __restrict__ denom, float* __restrict__ pbuf,
                            int heads) {
    int idx = blockIdx.x * blockDim.x + threadIdx.x;
    if (idx >= E_TOT * heads) return;
    int e = idx / heads, hd = idx - e * heads;
    int s, d; edge_sd(e, srcI, dstI, s, d);
    float v = es[s * heads + hd] + ed[d * heads + hd];
    v = v > 0.f ? v : NEG_SLOPE * v;
    float m = ord2f(mbits[d * heads + hd]);
    float p = __expf(v - m);
    pbuf[idx] = p;
    atomicAdd(&denom[d * heads + hd], p);
}

// ---- pass 4: out[dst] += (p/denom[dst]) * h[src] --------------------------
__global__ void edge_scatter(const int* __restrict__ srcI, const int* __restrict__ dstI,
                             const float* __restrict__ pbuf,
                             const float* __restrict__ denom,
                             const float* __restrict__ hW,
                             float* __restrict__ out, int heads, int C) {
    const int HHloc = heads * C;
    long long idx = (long long)blockIdx.x * blockDim.x + threadIdx.x;
    long long total = (long long)E_TOT * HHloc;
    if (idx >= total) return;
    int e = (int)(idx / HHloc);
    int c = (int)(idx - (long long)e * HHloc);
    int hd = c / C;
    int s, d; edge_sd(e, srcI, dstI, s, d);
    float alpha = pbuf[e * heads + hd] / denom[d * heads + hd];
    atomicAdd(&out[(size_t)d * HHloc + c], alpha * hW[(size_t)s * HHloc + c]);
}

// ---- bias + relu ----------------------------------------------------------
__global__ void bias_relu(float* __restrict__ h, const float* __restrict__ b,
                          int n, int HHloc) {
    int idx = blockIdx.x * blockDim.x + threadIdx.x;
    if (idx >= n * HHloc) return;
    float v = h[idx] + b[idx % HHloc];
    h[idx] = v > 0.f ? v : 0.f;
}

// ---- global mean pool accumulation ----------------------------------------
__global__ void pool_accum(const float* __restrict__ h, const int* __restrict__ batch,
                           float* __restrict__ gsum, float* __restrict__ gcnt) {
    int idx = blockIdx.x * blockDim.x + threadIdx.x;
    if (idx >= N_NODES * HH) return;
    int node = idx / HH, c = idx - node * HH;
    int g = batch[node];
    atomicAdd(&gsum[g * HH + c], h[idx]);
    if (c == 0) atomicAdd(&gcnt[g], 1.0f);
}

// ---- classifier head: sigmoid(pooled @ Wc + bc) ---------------------------
__global__ void final_head(const float* __restrict__ gsum, const float* __restrict__ gcnt,
                           const float* __restrict__ Wc, const float* __restrict__ bc,
                           float* __restrict__ out) {
    int g = blockIdx.x * blockDim.x + threadIdx.x;
    if (g >= N_GRAPHS) return;
    float cnt = gcnt[g];
    cnt = cnt > 1.f ? cnt : 1.f;
    float s = bc[0];
    for (int c = 0; c < HH; ++c)
        s += (gsum[g * HH + c] / cnt) * Wc[c];
    out[g] = 1.f / (1.f + __expf(-s));
}

// ---------------------------------------------------------------------------
// Host-side layer driver
// ---------------------------------------------------------------------------
static void gat_layer(hipStream_t stream,
                      const float* xin, int inC,
                      const float* W, const float* a_s, const float* a_d,
                      const float* bias, int heads, int C,
                      const int* srcI, const int* dstI,
                      float* hW, float* out,
                      float* es, float* ed, unsigned* mbits, float* denom,
                      float* pbuf) {
    const int HHloc = heads * C;   // always 96 here
    // hW = xin @ W
    dim3 gg(N_NODES / 16, HHloc / 16);
    gemm_wmma_f32<<<gg, 32, 0, stream>>>(xin, W, hW, N_NODES, HHloc, inC);
    // per-node logits
    {
        int t = N_NODES * heads;
        node_scores<<<(t + 255) / 256, 256, 0, stream>>>(hW, a_s, a_d, es, ed,
                                                         N_NODES, heads, C);
    }
    hipMemsetAsync(mbits, 0, (size_t)N_NODES * heads * sizeof(unsigned), stream);
    hipMemsetAsync(denom, 0, (size_t)N_NODES * heads * sizeof(float), stream);
    hipMemsetAsync(out, 0, (size_t)N_NODES * HHloc * sizeof(float), stream);
    {
        int t = E_TOT * heads;
        edge_max<<<(t + 255) / 256, 256, 0, stream>>>(srcI, dstI, es, ed, mbits, heads);
        edge_expsum<<<(t + 255) / 256, 256, 0, stream>>>(srcI, dstI, es, ed, mbits,
                                                         denom, pbuf, heads);
    }
    {
        long long t = (long long)E_TOT * HHloc;
        int blocks = (int)((t + 255) / 256);
        edge_scatter<<<blocks, 256, 0, stream>>>(srcI, dstI, pbuf, denom, hW, out,
                                                 heads, C);
    }
    {
        int t = N_NODES * HHloc;
        bias_relu<<<(t + 255) / 256, 256, 0, stream>>>(out, bias, N_NODES, HHloc);
    }
}

extern "C" void kernel_launch(void* const* d_in, const int* in_sizes, int n_in,
                              void* d_out, int out_size, void* d_ws, size_t ws_size,
                              hipStream_t stream) {
    // ---- inputs (setup_inputs dict order) ----
    const float* x    = (const float*)d_in[0];                 // [50000,128]
    const int*   ei   = (const int*)d_in[1];                   // [2,800000]
    // d_in[2] = edge_weight (ignored by GATConv)
    const int*   batch = (const int*)d_in[3];                  // [50000]
    const float* W1  = (const float*)d_in[4];
    const float* as1 = (const float*)d_in[5];
    const float* ad1 = (const float*)d_in[6];
    const float* b1  = (const float*)d_in[7];
    const float* W2  = (const float*)d_in[8];
    const float* as2 = (const float*)d_in[9];
    const float* ad2 = (const float*)d_in[10];
    const float* b2  = (const float*)d_in[11];
    const float* W3  = (const float*)d_in[12];
    const float* as3 = (const float*)d_in[13];
    const float* ad3 = (const float*)d_in[14];
    const float* b3  = (const float*)d_in[15];
    const float* W4  = (const float*)d_in[16];
    const float* as4 = (const float*)d_in[17];
    const float* ad4 = (const float*)d_in[18];
    const float* b4  = (const float*)d_in[19];
    const float* Wc  = (const float*)d_in[20];
    const float* bc  = (const float*)d_in[21];
    float* out = (float*)d_out;                                // [64]

    const int* srcI = ei;              // edge_index[0]
    const int* dstI = ei + N_EDGES;    // edge_index[1]

    // ---- workspace layout ----
    float* p = (float*)d_ws;
    float* hA = p;  p += (size_t)N_NODES * HH;    // GEMM result (reused each layer)
    float* hB = p;  p += (size_t)N_NODES * HH;    // ping
    float* hC = p;  p += (size_t)N_NODES * HH;    // pong
    float* es = p;  p += (size_t)N_NODES * HEADS;
    float* ed = p;  p += (size_t)N_NODES * HEADS;
    unsigned* mbits = (unsigned*)p; p += (size_t)N_NODES * HEADS;
    float* denom = p; p += (size_t)N_NODES * HEADS;
    float* pbuf = p;  p += (size_t)E_TOT * HEADS;
    float* gsum = p;  p += (size_t)N_GRAPHS * HH;
    float* gcnt = p;  p += (size_t)N_GRAPHS;
    (void)ws_size; (void)n_in; (void)in_sizes; (void)out_size;

    // Layer 1: x[50000,128] -> hB[50000,96], 2 heads x 48
    gat_layer(stream, x, IN_CH, W1, as1, ad1, b1, HEADS, HID,
              srcI, dstI, hA, hB, es, ed, mbits, denom, pbuf);
    // Layer 2: hB -> hC, 1 head x 96
    gat_layer(stream, hB, HH, W2, as2, ad2, b2, 1, HH,
              srcI, dstI, hA, hC, es, ed, mbits, denom, pbuf);
    // Layer 3: hC -> hB
    gat_layer(stream, hC, HH, W3, as3, ad3, b3, 1, HH,
              srcI, dstI, hA, hB, es, ed, mbits, denom, pbuf);
    // Layer 4: hB -> hC
    gat_layer(stream, hB, HH, W4, as4, ad4, b4, 1, HH,
              srcI, dstI, hA, hC, es, ed, mbits, denom, pbuf);

    // Global mean pool + sigmoid head
    hipMemsetAsync(gsum, 0, (size_t)N_GRAPHS * HH * sizeof(float), stream);
    hipMemsetAsync(gcnt, 0, (size_t)N_GRAPHS * sizeof(float), stream);
    {
        int t = N_NODES * HH;
        pool_accum<<<(t + 255) / 256, 256, 0, stream>>>(hC, batch, gsum, gcnt);
    }
    final_head<<<1, 64, 0, stream>>>(gsum, gcnt, Wc, bc, out);
}